// LinearModel_69114613728221
// MI455X (gfx1250) — compile-verified
//
#include <hip/hip_runtime.h>

// ---------------------------------------------------------------------------
// Reference collapse:  x_final = x @ (P^nTurns)^T,  P = W127 @ W126 @ ... @ W0
// Kernel 1: one block folds 128 6x6 fp32 matrices (tree product via
//           v_wmma_f32_16x16x4_f32, LDS ping-pong) then powers by nTurns.
// Kernel 2: streaming pass; each wave applies M to 16-particle tiles with
//           two chained fp32 WMMAs; B operand built once per wave; A operand
//           is two unconditional b64 loads per lane (k=6,7 slots cancelled by
//           zero B rows), so the hot loop has no EXEC divergence on loads.
// ---------------------------------------------------------------------------

typedef __attribute__((ext_vector_type(2))) float v2f;
typedef __attribute__((ext_vector_type(8))) float v8f;

#define N_ELEM 128
#define MATF   36        // 6*6 floats per matrix

// D(6x6) = A(6x6) @ B(6x6) via two V_WMMA_F32_16X16X4_F32 (K padded 6->8).
// A operand (16x4 fp32): lanes 0-15 -> M=lane, VGPR v -> K = 2*(lane>>4)+v.
// B operand (4x16 fp32): lanes 0-15 -> N=lane, VGPR v -> K = 2*(lane>>4)+v.
// D (16x16 fp32, 8 VGPRs): lane -> N = lane&15, VGPR r -> M = r + 8*(lane>>4).
__device__ __forceinline__ void mat6_mul_wmma(const float* __restrict__ A,
                                              const float* __restrict__ B,
                                              float* __restrict__ D,
                                              int lane) {
  const int half = lane >> 4;
  const int mn   = lane & 15;
  v2f a0, a1, b0, b1;
#pragma unroll
  for (int v = 0; v < 2; ++v) {
    const int k0 = 2 * half + v;       // 0..3  (always < 6)
    const int k1 = 4 + 2 * half + v;   // 4..7  (6,7 are zero padding)
    a0[v] = (mn < 6)            ? A[mn * 6 + k0] : 0.0f;
    a1[v] = (mn < 6 && k1 < 6)  ? A[mn * 6 + k1] : 0.0f;
    b0[v] = (mn < 6)            ? B[k0 * 6 + mn] : 0.0f;
    b1[v] = (mn < 6 && k1 < 6)  ? B[k1 * 6 + mn] : 0.0f;
  }
  v8f c = {};
  c = __builtin_amdgcn_wmma_f32_16x16x4_f32(false, a0, false, b0,
                                            (short)0, c, false, false);
  c = __builtin_amdgcn_wmma_f32_16x16x4_f32(false, a1, false, b1,
                                            (short)0, c, false, false);
  if (half == 0 && mn < 6) {
#pragma unroll
    for (int r = 0; r < 6; ++r) D[r * 6 + mn] = c[r];  // rows 0..5, col mn
  }
}

// One block, 256 threads (8 waves). Computes M = (W127@...@W0)^nTurns -> Mout[36].
__global__ void __launch_bounds__(256)
chain_pow_kernel(const float* __restrict__ W,
                 const int* __restrict__ nTurnsPtr,
                 float* __restrict__ Mout) {
  __shared__ float buf[(N_ELEM + N_ELEM / 2) * MATF];  // ping-pong regions
  const int tid  = threadIdx.x;
  const int lane = tid & 31;
  const int wave = tid >> 5;

  // Load all 128 matrices (18 KB) coalesced into LDS.
  for (int i = tid; i < N_ELEM * MATF; i += 256) buf[i] = W[i];
  __syncthreads();

  // Tree product: level pairs src[2p+1] @ src[2p]  (left factor = later element).
  int srcOff = 0, dstOff = N_ELEM * MATF;
  int count = N_ELEM;
  while (count > 1) {
    const int nprod = count >> 1;
    for (int p = wave; p < nprod; p += 8)
      mat6_mul_wmma(&buf[srcOff + (2 * p + 1) * MATF],
                    &buf[srcOff + (2 * p) * MATF],
                    &buf[dstOff + p * MATF], lane);
    __syncthreads();
    const int t = srcOff; srcOff = dstOff; dstOff = t;
    count = nprod;
  }
  // P now at buf[srcOff]. Compute M = P^nTurns (ping-pong in dst region).
  const int nT = *nTurnsPtr;
  int mOff = dstOff, tOff = dstOff + MATF;
  if (tid < MATF) buf[mOff + tid] = ((tid % 7) == 0) ? 1.0f : 0.0f;  // M = I
  __syncthreads();
  for (int t = 0; t < nT; ++t) {
    if (wave == 0)
      mat6_mul_wmma(&buf[srcOff], &buf[mOff], &buf[tOff], lane);
    __syncthreads();
    const int s = mOff; mOff = tOff; tOff = s;
  }
  if (tid < MATF) Mout[tid] = buf[mOff + tid];
}

// Streaming apply: x_new[p][n] = sum_k x[p][k] * M[n][k].
__global__ void __launch_bounds__(256)
apply_kernel(const float* __restrict__ X,
             const float* __restrict__ M,
             float* __restrict__ Y,
             int nTiles) {
  const int lane   = threadIdx.x & 31;
  const int gwave  = (blockIdx.x * blockDim.x + threadIdx.x) >> 5;
  const int nWaves = (gridDim.x * blockDim.x) >> 5;
  const int half   = lane >> 4;
  const int mn     = lane & 15;

  // B[k][n] = M[n][k]; pad n>=6 and k>=6 with zeros (one-time per wave).
  v2f b0, b1;
#pragma unroll
  for (int v = 0; v < 2; ++v) {
    const int k0 = 2 * half + v;       // 0..3
    const int k1 = 4 + 2 * half + v;   // 4..7
    b0[v] = (mn < 6)           ? M[mn * 6 + k0] : 0.0f;
    b1[v] = (mn < 6 && k1 < 6) ? M[mn * 6 + k1] : 0.0f;
  }

  for (int tile = gwave; tile < nTiles; tile += nWaves) {
    const float* row = X + (size_t)(tile * 16 + mn) * 6;  // this lane's particle
    // A operand, branch-free:
    //  a0 holds K = 2*half, 2*half+1  (8B-aligned b64 load)
    //  a1 holds K = 4,5 for ALL lanes; half=1 lanes map to K=6,7 whose B rows
    //  are zero, so the loaded values are cancelled. 8B-aligned b64 load.
    v2f a0 = *(const v2f*)(row + 2 * half);
    v2f a1 = *(const v2f*)(row + 4);
    v8f c = {};
    c = __builtin_amdgcn_wmma_f32_16x16x4_f32(false, a0, false, b0,
                                              (short)0, c, false, false);
    c = __builtin_amdgcn_wmma_f32_16x16x4_f32(false, a1, false, b1,
                                              (short)0, c, false, false);
    if (mn < 6) {
#pragma unroll
      for (int r = 0; r < 8; ++r) {
        const int m = r + 8 * half;                  // output row in tile
        Y[(size_t)(tile * 16 + m) * 6 + mn] = c[r];
      }
    }
  }
}

extern "C" void kernel_launch(void* const* d_in, const int* in_sizes, int n_in,
                              void* d_out, int out_size, void* d_ws, size_t ws_size,
                              hipStream_t stream) {
  const float* X  = (const float*)d_in[0];   // (131072, 6) fp32
  const float* W  = (const float*)d_in[1];   // (128, 6, 6) fp32
  const int*   nT = (const int*)d_in[2];     // scalar nTurns
  float* Y    = (float*)d_out;               // (131072, 6) fp32
  float* Mtot = (float*)d_ws;                // 36 floats scratch

  chain_pow_kernel<<<1, 256, 0, stream>>>(W, nT, Mtot);

  const int nPart  = in_sizes[0] / 6;
  const int nTiles = nPart / 16;             // 8192 tiles of 16 particles
  apply_kernel<<<512, 256, 0, stream>>>(X, Mtot, Y, nTiles);
}